// InteractionBlock_11940009083651
// MI455X (gfx1250) — compile-verified
//
#include <hip/hip_runtime.h>

// ---------------------------------------------------------------------------
// SchNet InteractionBlock for MI455X (gfx1250, wave32, WMMA, async-to-LDS).
//
//  W  = ssp(edge_attr @ nn_w1 + b1) @ nn_w2 + b2        [E, 128]
//  C  = cosine cutoff(edge_length);  W *= C
//  h  = x @ lin1_w                                      [N, 128]
//  agg[dst] += h[src] * W        (fp32 global atomics)
//  out = ssp(agg @ lin2_w + b) @ lin_w + b              [N, 128]
//
// fp32-exact GEMMs via V_WMMA_F32_16X16X4_F32. Weight B-fragments are held
// in VGPRs (loop-invariant per wave). A-fragments are burst-loaded from LDS
// into unrolled register arrays (16 k-steps at a time) so the DS loads
// pipeline and the WMMAs issue back-to-back (accumulator chaining D->C has
// no hazard penalty). Tile staging in the edge kernel uses
// GLOBAL_LOAD_ASYNC_TO_LDS_B128 tracked by ASYNCcnt.
// ---------------------------------------------------------------------------

#define N_NODES 50000
#define E_EDGES 1600000
#define HID     128
#define NF      128
#define NG      64

#define EASTR   68    // padded LDS stride for 64-wide tiles  (bank-conflict free)
#define MSTR    132   // padded LDS stride for 128-wide tiles (bank-conflict free)

typedef __attribute__((ext_vector_type(2))) float v2f;
typedef __attribute__((ext_vector_type(8))) float v8f;

// D = A(16x4) * B(4x16) + C, fp32, wave32.
// A layout: lane m=lane&15, ks=(lane>>4)*2 -> a.x=A[m][k0+ks], a.y=A[m][k0+ks+1]
// B layout: lane n=lane&15, ks=(lane>>4)*2 -> b.x=B[k0+ks][n], b.y=B[k0+ks+1][n]
// C/D: vgpr r -> row = r + 8*(lane>>4), col = lane&15
__device__ __forceinline__ v8f wmma_f32_k4(v2f a, v2f b, v8f c) {
  return __builtin_amdgcn_wmma_f32_16x16x4_f32(
      /*neg_a=*/false, a, /*neg_b=*/false, b,
      /*c_mod=*/(short)0, c, /*reuse_a=*/false, /*reuse_b=*/false);
}

// ShiftedSoftplus: logaddexp(v,0) - log(2), numerically stable.
__device__ __forceinline__ float sspf(float v) {
  return fmaxf(v, 0.0f) + log1pf(__expf(-fabsf(v))) - 0.69314718055994531f;
}

// LDS byte offset of a __shared__ object (generic-pointer low 32 bits map to
// the LDS aperture offset on gfx1250).
__device__ __forceinline__ unsigned lds_off(const void* p) {
  return (unsigned)(size_t)p;
}

// Async copy 16B global -> LDS (ASYNCcnt-tracked, no VGPR return traffic).
__device__ __forceinline__ void async_b128(unsigned lds, const void* gaddr) {
  asm volatile("global_load_async_to_lds_b128 %0, %1, off"
               :: "v"(lds), "v"(gaddr) : "memory");
}
__device__ __forceinline__ void async_wait0() {
  asm volatile("s_wait_asynccnt 0" ::: "memory");
}

// Burst-load 16 A-fragments (one 16-step K chunk) from an LDS tile row.
// base points at &tile[l16 * stride]; k0 is the first k of the chunk.
__device__ __forceinline__ void load_afrag16(v2f af[16], const float* base,
                                             int k0, int ks) {
#pragma unroll
  for (int kk = 0; kk < 16; ++kk) {
    af[kk].x = base[k0 + kk * 4 + ks];
    af[kk].y = base[k0 + kk * 4 + ks + 1];
  }
}

// ---------------------------------------------------------------------------
__global__ void k_zero(float* __restrict__ p, long n) {
  long stride = (long)gridDim.x * blockDim.x * 4;
  for (long j = ((long)blockIdx.x * blockDim.x + threadIdx.x) * 4; j < n; j += stride) {
    float4 z = make_float4(0.f, 0.f, 0.f, 0.f);
    *(float4*)(p + j) = z;
  }
}

// ---------------------------------------------------------------------------
// h = x @ lin1_w    ([N,128] @ [128,128]); B fragments registerized (64 VGPRs)
__global__ __launch_bounds__(256) void k_gemm_h(const float* __restrict__ x,
                                                const float* __restrict__ w,
                                                float* __restrict__ h,
                                                int ntiles) {
  __shared__ float sX[16 * MSTR];

  const int tid  = threadIdx.x;
  const int lane = tid & 31, wave = tid >> 5;
  const int hlf = lane >> 4, l16 = lane & 15;
  const int ks = hlf * 2, colBase = wave * 16;

  v2f bf[32];                         // lin1_w fragments, loop-invariant
#pragma unroll
  for (int kk = 0; kk < 32; ++kk) {
    const int k = kk * 4 + ks;
    bf[kk].x = w[(k)     * NF + colBase + l16];
    bf[kk].y = w[(k + 1) * NF + colBase + l16];
  }

  for (int t = blockIdx.x; t < ntiles; t += gridDim.x) {
    const int row0 = t * 16;
    {  // stage 16x128 x-tile
      const int row = tid >> 4, cb = (tid & 15) * 8;
      const float* xp = x + (size_t)(row0 + row) * HID + cb;
      *(float4*)(sX + row * MSTR + cb)     = *(const float4*)(xp);
      *(float4*)(sX + row * MSTR + cb + 4) = *(const float4*)(xp + 4);
    }
    __syncthreads();

    const float* arow = sX + l16 * MSTR;
    v8f c = {};
#pragma unroll
    for (int half = 0; half < 2; ++half) {      // K = 128 as 2 x 16-step bursts
      v2f af[16];
      load_afrag16(af, arow, half * 64, ks);
#pragma unroll
      for (int kk = 0; kk < 16; ++kk)
        c = wmma_f32_k4(af[kk], bf[half * 16 + kk], c);
    }
#pragma unroll
    for (int r = 0; r < 8; ++r) {
      const int row = r + 8 * hlf;
      h[(size_t)(row0 + row) * NF + colBase + l16] = c[r];
    }
    __syncthreads();
  }
}

// ---------------------------------------------------------------------------
// Edge kernel: filter MLP + cutoff envelope + gather h[src] + scatter-add.
// One workgroup = 8 waves; each wave owns one 16-col slab of the 128 features.
// nn_w1/nn_w2 fragments live in VGPRs (32 + 64); tiles staged via async-to-LDS.
__global__ __launch_bounds__(256) void k_edge(const float* __restrict__ edge_attr,
                                              const float* __restrict__ edge_length,
                                              const int*   __restrict__ edge_index,
                                              const float* __restrict__ nn_w1,
                                              const float* __restrict__ nn_b1,
                                              const float* __restrict__ nn_w2,
                                              const float* __restrict__ nn_b2,
                                              const float* __restrict__ h,
                                              float*       __restrict__ agg,
                                              int ntiles) {
  __shared__ float sEA[16 * EASTR];   // edge_attr tile  [16 x 64]
  __shared__ float sMid[16 * MSTR];   // ssp(EA@W1+b1)   [16 x 128]
  __shared__ float sH[16 * MSTR];     // gathered h[src] [16 x 128]
  __shared__ float sEnv[16];
  __shared__ int   sDst[16];

  const int tid  = threadIdx.x;
  const int lane = tid & 31, wave = tid >> 5;
  const int hlf = lane >> 4, l16 = lane & 15;
  const int ks = hlf * 2, colBase = wave * 16;
  const float b1 = nn_b1[colBase + l16];
  const float b2 = nn_b2[colBase + l16];

  v2f bf1[16];                        // nn_w1 fragments (K=64)
#pragma unroll
  for (int kk = 0; kk < 16; ++kk) {
    const int k = kk * 4 + ks;
    bf1[kk].x = nn_w1[(k)     * NF + colBase + l16];
    bf1[kk].y = nn_w1[(k + 1) * NF + colBase + l16];
  }
  v2f bf2[32];                        // nn_w2 fragments (K=128)
#pragma unroll
  for (int kk = 0; kk < 32; ++kk) {
    const int k = kk * 4 + ks;
    bf2[kk].x = nn_w2[(k)     * NF + colBase + l16];
    bf2[kk].y = nn_w2[(k + 1) * NF + colBase + l16];
  }

  // per-thread staging coordinates
  const int srow = tid >> 4;           // 0..15
  const int cb4  = (tid & 15) * 4;     // edge_attr tile cols
  const int cb8  = (tid & 15) * 8;     // h tile cols

  for (int t = blockIdx.x; t < ntiles; t += gridDim.x) {
    const int e0 = t * 16;

    // --- async stage edge_attr tile [16 x 64]: one b128 per thread ---
    async_b128(lds_off(sEA + srow * EASTR + cb4),
               edge_attr + (size_t)e0 * NG + srow * NG + cb4);

    // --- async gather h[src] rows [16 x 128]: two b128 per thread ---
    const int srcn = edge_index[e0 + srow];
    const float* hp = h + (size_t)srcn * NF + cb8;
    async_b128(lds_off(sH + srow * MSTR + cb8),     hp);
    async_b128(lds_off(sH + srow * MSTR + cb8 + 4), hp + 4);

    if (tid < 16) {  // cutoff envelope + destination indices
      const float len = edge_length[e0 + tid];
      float cc = 0.5f * (__cosf(len * 0.31415926535897931f) + 1.0f);
      cc = (len <= 10.0f && len >= 0.0f) ? cc : 0.0f;
      sEnv[tid] = cc;
      sDst[tid] = edge_index[E_EDGES + e0 + tid];
    }

    // prefetch next tile's edge_attr while WMMAs run
    if (t + (int)gridDim.x < ntiles) {
      const int en = (t + gridDim.x) * 16;
      __builtin_prefetch(edge_attr + (size_t)en * NG + srow * NG + cb4, 0, 0);
    }

    async_wait0();       // this wave's async LDS writes have landed
    __syncthreads();     // all waves' writes visible

    // GEMM1: mid = ssp(EA @ W1 + b1), K = 64 (one 16-step burst)
    v8f c1 = {};
    {
      v2f af[16];
      load_afrag16(af, sEA + l16 * EASTR, 0, ks);
#pragma unroll
      for (int kk = 0; kk < 16; ++kk)
        c1 = wmma_f32_k4(af[kk], bf1[kk], c1);
    }
#pragma unroll
    for (int r = 0; r < 8; ++r) {
      const int row = r + 8 * hlf;
      sMid[row * MSTR + colBase + l16] = sspf(c1[r] + b1);
    }
    __syncthreads();

    // GEMM2: W = mid @ W2 + b2, K = 128 (2 x 16-step bursts)
    v8f c2 = {};
#pragma unroll
    for (int half = 0; half < 2; ++half) {
      v2f af[16];
      load_afrag16(af, sMid + l16 * MSTR, half * 64, ks);
#pragma unroll
      for (int kk = 0; kk < 16; ++kk)
        c2 = wmma_f32_k4(af[kk], bf2[half * 16 + kk], c2);
    }
    // msg = (W + b2) * env * h[src]; scatter-add to agg[dst]
#pragma unroll
    for (int r = 0; r < 8; ++r) {
      const int row = r + 8 * hlf;
      const float wv  = (c2[r] + b2) * sEnv[row];
      const float msg = wv * sH[row * MSTR + colBase + l16];
      atomicAdd(&agg[(size_t)sDst[row] * NF + colBase + l16], msg);
    }
    __syncthreads();
  }
}

// ---------------------------------------------------------------------------
// out = ssp(agg @ lin2_w + lin2_b) @ lin_w + lin_b; both weight sets in VGPRs.
__global__ __launch_bounds__(256) void k_out(const float* __restrict__ agg,
                                             const float* __restrict__ w2,
                                             const float* __restrict__ b2v,
                                             const float* __restrict__ w3,
                                             const float* __restrict__ b3v,
                                             float* __restrict__ out,
                                             int ntiles) {
  __shared__ float sA[16 * MSTR];
  __shared__ float sT[16 * MSTR];

  const int tid  = threadIdx.x;
  const int lane = tid & 31, wave = tid >> 5;
  const int hlf = lane >> 4, l16 = lane & 15;
  const int ks = hlf * 2, colBase = wave * 16;
  const float bias2 = b2v[colBase + l16];
  const float bias3 = b3v[colBase + l16];

  v2f bfA[32], bfB[32];
#pragma unroll
  for (int kk = 0; kk < 32; ++kk) {
    const int k = kk * 4 + ks;
    bfA[kk].x = w2[(k)     * HID + colBase + l16];
    bfA[kk].y = w2[(k + 1) * HID + colBase + l16];
    bfB[kk].x = w3[(k)     * HID + colBase + l16];
    bfB[kk].y = w3[(k + 1) * HID + colBase + l16];
  }

  for (int t = blockIdx.x; t < ntiles; t += gridDim.x) {
    const int row0 = t * 16;
    {  // stage agg tile
      const int row = tid >> 4, cb = (tid & 15) * 8;
      const float* ap = agg + (size_t)(row0 + row) * NF + cb;
      *(float4*)(sA + row * MSTR + cb)     = *(const float4*)(ap);
      *(float4*)(sA + row * MSTR + cb + 4) = *(const float4*)(ap + 4);
    }
    __syncthreads();

    // GEMM1: ssp(agg @ lin2_w + b), 2 x 16-step bursts
    v8f c1 = {};
#pragma unroll
    for (int half = 0; half < 2; ++half) {
      v2f af[16];
      load_afrag16(af, sA + l16 * MSTR, half * 64, ks);
#pragma unroll
      for (int kk = 0; kk < 16; ++kk)
        c1 = wmma_f32_k4(af[kk], bfA[half * 16 + kk], c1);
    }
#pragma unroll
    for (int r = 0; r < 8; ++r) {
      const int row = r + 8 * hlf;
      sT[row * MSTR + colBase + l16] = sspf(c1[r] + bias2);
    }
    __syncthreads();

    // GEMM2: @ lin_w + b, 2 x 16-step bursts
    v8f c2 = {};
#pragma unroll
    for (int half = 0; half < 2; ++half) {
      v2f af[16];
      load_afrag16(af, sT + l16 * MSTR, half * 64, ks);
#pragma unroll
      for (int kk = 0; kk < 16; ++kk)
        c2 = wmma_f32_k4(af[kk], bfB[half * 16 + kk], c2);
    }
#pragma unroll
    for (int r = 0; r < 8; ++r) {
      const int row = r + 8 * hlf;
      out[(size_t)(row0 + row) * HID + colBase + l16] = c2[r] + bias3;
    }
    __syncthreads();
  }
}

// ---------------------------------------------------------------------------
extern "C" void kernel_launch(void* const* d_in, const int* in_sizes, int n_in,
                              void* d_out, int out_size, void* d_ws, size_t ws_size,
                              hipStream_t stream) {
  const float* x           = (const float*)d_in[0];
  const int*   edge_index  = (const int*)  d_in[1];
  const float* edge_length = (const float*)d_in[2];
  const float* edge_attr   = (const float*)d_in[3];
  const float* lin1_w      = (const float*)d_in[4];
  const float* nn_w1       = (const float*)d_in[5];
  const float* nn_b1       = (const float*)d_in[6];
  const float* nn_w2       = (const float*)d_in[7];
  const float* nn_b2       = (const float*)d_in[8];
  const float* lin2_w      = (const float*)d_in[9];
  const float* lin2_b      = (const float*)d_in[10];
  const float* lin_w       = (const float*)d_in[11];
  const float* lin_b       = (const float*)d_in[12];
  float* out = (float*)d_out;

  float* h   = (float*)d_ws;                  // [N, NF]
  float* agg = h + (size_t)N_NODES * NF;      // [N, NF]

  const int node_tiles = N_NODES / 16;        // 3125 (exact)
  const int edge_tiles = E_EDGES / 16;        // 100000 (exact)

  k_zero<<<1024, 256, 0, stream>>>(agg, (long)N_NODES * NF);
  k_gemm_h<<<node_tiles, 256, 0, stream>>>(x, lin1_w, h, node_tiles);
  k_edge<<<6250, 256, 0, stream>>>(edge_attr, edge_length, edge_index,
                                   nn_w1, nn_b1, nn_w2, nn_b2, h, agg, edge_tiles);
  k_out<<<625, 256, 0, stream>>>(agg, lin2_w, lin2_b, lin_w, lin_b, out, node_tiles);
}